// PredictLoss_88115549045582
// MI455X (gfx1250) — compile-verified
//
#include <hip/hip_runtime.h>
#include <hip/hip_bf16.h>

// Problem constants from the reference (B, N, D) = (8, 4096, 64)
#define BATCH 8
#define NROWS 4096
#define DDIM  64
#define NCHUNKS 32            // N / 128 row-chunks for the t reduction
#define ROWS_PER_CHUNK 128

typedef __attribute__((ext_vector_type(2))) float v2f;
typedef __attribute__((ext_vector_type(8))) float v8f;

// ---------------------------------------------------------------------------
// K1: per-row inverse L2 norm.  rinv[b*N+n] = 1/sqrt(sum_d x^2)
// one thread per row; 16x float4 loads (256 B per row)
// ---------------------------------------------------------------------------
__global__ void k1_row_rinv(const float* __restrict__ x, float* __restrict__ rinv) {
    const int row = blockIdx.x * blockDim.x + threadIdx.x;   // 0 .. B*N-1
    const float4* rp = (const float4*)(x + (size_t)row * DDIM);
    float ss = 0.0f;
#pragma unroll
    for (int i = 0; i < DDIM / 4; ++i) {
        float4 v = rp[i];
        ss += v.x * v.x + v.y * v.y + v.z * v.z + v.w * v.w;
    }
    rinv[row] = 1.0f / sqrtf(ss);
}

// ---------------------------------------------------------------------------
// K2: per-chunk column partial sums of the normalized rows.
// partial[b][chunk][d] = sum_{n in chunk} x[b,n,d] * rinv[b,n]
// block = (b, chunk of 128 rows); tid -> d = tid&63 (coalesced), g = tid>>6
// Deterministic: fixed iteration order, fixed LDS reduce order.
// ---------------------------------------------------------------------------
__global__ void k2_col_partials(const float* __restrict__ x,
                                const float* __restrict__ rinv,
                                float* __restrict__ partial) {
    const int b     = blockIdx.x >> 5;
    const int chunk = blockIdx.x & 31;
    const int d     = threadIdx.x & 63;
    const int g     = threadIdx.x >> 6;          // 0..3 row groups
    __shared__ float red[4][DDIM];

    const size_t base = ((size_t)b * NROWS + (size_t)chunk * ROWS_PER_CHUNK);
    float acc = 0.0f;
#pragma unroll 4
    for (int i = 0; i < ROWS_PER_CHUNK / 4; ++i) {
        const int n = 4 * i + g;
        acc += x[(base + n) * DDIM + d] * rinv[base + n];
    }
    red[g][d] = acc;
    __syncthreads();
    if (threadIdx.x < DDIM) {
        float s = red[0][d] + red[1][d] + red[2][d] + red[3][d]; // fixed order
        partial[((size_t)b * NCHUNKS + chunk) * DDIM + d] = s;
    }
}

// ---------------------------------------------------------------------------
// K3: reduce chunk partials -> t[b][d]   (grid = B, block = 64, fixed order)
// ---------------------------------------------------------------------------
__global__ void k3_reduce_t(const float* __restrict__ partial, float* __restrict__ t) {
    const int b = blockIdx.x;
    const int d = threadIdx.x;
    float s = 0.0f;
    for (int c = 0; c < NCHUNKS; ++c)
        s += partial[((size_t)b * NCHUNKS + c) * DDIM + d];
    t[b * DDIM + d] = s;
}

// ---------------------------------------------------------------------------
// K4: loss[b,m] = (t_b . s[m] - 1) / (N-1) via V_WMMA_F32_16X16X4_F32.
// One wave handles a 16-row tile; A = rinv-scaled row tile (16x4 per step),
// B = t broadcast into all 16 columns; 16 chained WMMAs accumulate K=64.
// A layout (ISA 7.12.2): lane L, M=L%16, h=L/16 holds A[M][k0+2h], A[M][k0+2h+1].
// C/D layout: VGPR r, lane-half h -> row 8h+r (all columns identical here).
// ---------------------------------------------------------------------------
__global__ void k4_wmma_dot(const float* __restrict__ x,
                            const float* __restrict__ rinv,
                            const float* __restrict__ t,
                            float* __restrict__ out) {
    const int b        = blockIdx.x >> 5;           // 8 batches
    const int tileset  = blockIdx.x & 31;           // 32 tilesets of 128 rows
    const int wave     = threadIdx.x >> 5;          // 8 waves/block
    const int lane     = threadIdx.x & 31;
    const int tile     = tileset * 128 + wave * 16; // first row of this wave's tile

    __shared__ float ts[DDIM];
    if (threadIdx.x < DDIM) ts[threadIdx.x] = t[b * DDIM + threadIdx.x];
    __syncthreads();

    const int h = lane >> 4;                        // lane half selects K pair
    const int m = tile + (lane & 15);               // matrix row owned by lane
    const float* rowp = x + ((size_t)b * NROWS + m) * DDIM;
    const float rm = rinv[(size_t)b * NROWS + m];

    v8f c = {};
#pragma unroll
    for (int k0 = 0; k0 < DDIM; k0 += 4) {
        const int base = k0 + 2 * h;
        float2 av = *(const float2*)(rowp + base);
        v2f a; a.x = av.x * rm; a.y = av.y * rm;    // normalized A tile
        v2f bb; bb.x = ts[base]; bb.y = ts[base + 1]; // t broadcast over columns
        c = __builtin_amdgcn_wmma_f32_16x16x4_f32(
                /*neg_a=*/false, a, /*neg_b=*/false, bb,
                /*c_mod=*/(short)0, c, /*reuse_a=*/false, /*reuse_b=*/false);
    }

    // every column of D is identical; lanes 0 and 16 extract rows 0..7 / 8..15
    if ((lane & 15) == 0) {
        const float scale = 1.0f / (float)(NROWS - 1);
        float* op = out + (size_t)b * NROWS + tile + 8 * h;
#pragma unroll
        for (int r = 0; r < 8; ++r)
            op[r] = (c[r] - 1.0f) * scale;          // s.s == 1 after normalize
    }
}

// ---------------------------------------------------------------------------
extern "C" void kernel_launch(void* const* d_in, const int* in_sizes, int n_in,
                              void* d_out, int out_size, void* d_ws, size_t ws_size,
                              hipStream_t stream) {
    (void)in_sizes; (void)n_in; (void)out_size; (void)ws_size;
    const float* x = (const float*)d_in[0];
    float* out = (float*)d_out;

    // scratch layout (floats): rinv[B*N] | partial[B*NCHUNKS*D] | t[B*D]
    float* rinv    = (float*)d_ws;
    float* partial = rinv + (size_t)BATCH * NROWS;
    float* tvec    = partial + (size_t)BATCH * NCHUNKS * DDIM;

    k1_row_rinv   <<<(BATCH * NROWS) / 256, 256, 0, stream>>>(x, rinv);
    k2_col_partials<<<BATCH * NCHUNKS,      256, 0, stream>>>(x, rinv, partial);
    k3_reduce_t   <<<BATCH,                  64, 0, stream>>>(partial, tvec);
    k4_wmma_dot   <<<BATCH * NCHUNKS,       256, 0, stream>>>(x, rinv, tvec, out);
}